// WinCorrTorch_Conv_81647328297402
// MI455X (gfx1250) — compile-verified
//
#include <hip/hip_runtime.h>

typedef __attribute__((ext_vector_type(2))) float v2f;
typedef __attribute__((ext_vector_type(8))) float v8f;

#define DIM       96
#define HWSZ      (96*96)
#define DHW       (96*96*96)
#define CSTRIDE   104              // padded w-stride per channel in LDS (floats)
#define ROWSTRIDE (32*CSTRIDE)     // one moving row (32 channels) = 3328 floats
#define LDSF      (3*ROWSTRIDE)    // 3 rows (hm = h-1..h+1) = 9984 floats ~ 39KB
#define WOFF      4                // real w data at LDS index w+WOFF (16B aligned)

// One 192-thread block (6 waves) = one (d,h) output row; wave w = w-tile.
// Per dm-plane i: block stages 3 moving rows into LDS (b128 global loads,
// zero-padded rows/halos written as zeros), then each wave runs 9 (j,k)
// correlation chains of 8x v_wmma_f32_16x16x4_f32 reading B from LDS with
// immediate-offset ds loads (no masking, no per-load address math).
__global__ __launch_bounds__(192)
void wincorr_wmma_kernel(const float* __restrict__ fixedp,
                         const float* __restrict__ movingp,
                         float* __restrict__ outp)
{
    __shared__ float lds[LDSF];

    const int t    = threadIdx.x;        // 0..191
    const int lane = t & 31;
    const int wv   = t >> 5;             // wave in block = w-tile 0..5
    const int h    = blockIdx.x % DIM;
    const int d    = blockIdx.x / DIM;
    const int w0   = wv * 16;

    // WMMA 32-bit A/B per-lane layout: spatial = lane&15, K = (lane>>4)*2 + vgpr
    const int sp    = lane & 15;
    const int khalf = (lane >> 4) * 2;

    // ---- A (fixed) tile: 8 K-slices of 4 channels, loaded once ----
    v2f a[8];
    const size_t fbase = (size_t)(d * HWSZ + h * DIM + w0 + sp);
    #pragma unroll
    for (int s = 0; s < 8; ++s) {
        const int c0 = 4 * s + khalf;
        a[s].x = fixedp[(size_t)c0       * DHW + fbase];
        a[s].y = fixedp[(size_t)(c0 + 1) * DHW + fbase];
    }

    // diagonal D[m,m]: lane m (vgpr m) for m<8, lane m+16 (vgpr m-8) for m>=8
    const bool diagLane = (lane < 8) || (lane >= 24);
    const int  didx     = (lane < 8) ? lane : (lane - 24);
    const int  m        = (lane < 16) ? lane : (lane - 16);
    const float  scale  = 0.17677669529663687f;   // 32^-0.5
    const size_t obase  = (size_t)(d * HWSZ + h * DIM + w0 + m);

    // ---- zero the halo columns once (w=-1 -> idx 3, w=96 -> idx 100) ----
    // 3 rows * 32 ch * 2 sides = 192 slots = exactly one per thread
    {
        const int hj = t >> 6, hc = (t & 63) >> 1, hs = t & 1;
        lds[hj * ROWSTRIDE + hc * CSTRIDE + (hs ? (WOFF + DIM) : (WOFF - 1))] = 0.0f;
    }

    for (int i = 0; i < 3; ++i) {
        const int dm = d + i - 1;
        const int dmc = dm < 0 ? 0 : (dm > DIM - 1 ? DIM - 1 : dm);

        if (i) __syncthreads();          // previous compute done before restage

        // ---- stage 3 moving rows (hm = h-1..h+1) for this dm ----
        // 3*32*24 = 2304 float4 segments, 12 per thread
        #pragma unroll
        for (int q = 0; q < 12; ++q) {
            const int seg = t + 192 * q;           // 0..2303
            const int j   = seg / 768;
            const int rem = seg % 768;
            const int c   = rem / 24;
            const int wq  = (rem % 24) * 4;
            const int hm  = h + j - 1;
            const bool ok = (dm >= 0) & (dm < DIM) & (hm >= 0) & (hm < DIM);
            const int hmc = hm < 0 ? 0 : (hm > DIM - 1 ? DIM - 1 : hm);
            const float4 src = *(const float4*)(movingp + (size_t)c * DHW +
                                    (size_t)(dmc * HWSZ + hmc * DIM + wq));
            float4 val;
            val.x = ok ? src.x : 0.0f;
            val.y = ok ? src.y : 0.0f;
            val.z = ok ? src.z : 0.0f;
            val.w = ok ? src.w : 0.0f;
            *(float4*)(lds + j * ROWSTRIDE + c * CSTRIDE + WOFF + wq) = val;
        }
        __syncthreads();

        // ---- 9 correlation chains (j,k) against LDS-resident rows ----
        #pragma unroll
        for (int j = 0; j < 3; ++j) {
          #pragma unroll
          for (int k = 0; k < 3; ++k) {
            // base for this lane: row j, channel khalf, voxel w0+sp+(k-1)
            const float* bp = lds + j * ROWSTRIDE + khalf * CSTRIDE
                                  + (WOFF - 1) + w0 + sp + k;
            v2f b[8];
            #pragma unroll
            for (int s = 0; s < 8; ++s) {
                b[s].x = bp[s * 4 * CSTRIDE];              // channel 4s+khalf
                b[s].y = bp[s * 4 * CSTRIDE + CSTRIDE];    // channel 4s+khalf+1
            }

            v8f acc = {};
            #pragma unroll
            for (int s = 0; s < 8; ++s) {
                acc = __builtin_amdgcn_wmma_f32_16x16x4_f32(
                          false, a[s], false, b[s], (short)0, acc, false, false);
            }

            // extract diagonal element (cndmask chain over the 8 acc VGPRs)
            float v = acc[0];
            if (didx == 1) v = acc[1];
            if (didx == 2) v = acc[2];
            if (didx == 3) v = acc[3];
            if (didx == 4) v = acc[4];
            if (didx == 5) v = acc[5];
            if (didx == 6) v = acc[6];
            if (didx == 7) v = acc[7];

            if (diagLane) {
                const int o = (i * 3 + j) * 3 + k;
                outp[(size_t)o * DHW + obase] = v * scale;
            }
          }
        }
    }
}

extern "C" void kernel_launch(void* const* d_in, const int* in_sizes, int n_in,
                              void* d_out, int out_size, void* d_ws, size_t ws_size,
                              hipStream_t stream)
{
    const float* fixedp  = (const float*)d_in[0];
    const float* movingp = (const float*)d_in[1];
    float* outp = (float*)d_out;

    // one block per (d,h) row: 96*96 = 9216 blocks of 192 threads (6 waves)
    dim3 grid(DIM * DIM), block(192);
    hipLaunchKernelGGL(wincorr_wmma_kernel, grid, block, 0, stream,
                       fixedp, movingp, outp);
}